// GCN_2800318677178
// MI455X (gfx1250) — compile-verified
//
#include <hip/hip_runtime.h>

typedef __attribute__((ext_vector_type(2))) float v2f;
typedef __attribute__((ext_vector_type(8))) float v8f;

#define GCN_N    100000
#define GCN_E    1600000
#define GCN_F    64      // IN_F == HID_F == 64
#define GCN_OUTF 32

// ---------------------------------------------------------------------------
// Zero-fill scratch (atomic accumulators must start at 0 every call).
// ---------------------------------------------------------------------------
__global__ void gcn_zero_kernel(float* __restrict__ p, long n) {
  long i = (long)blockIdx.x * blockDim.x + threadIdx.x;
  long stride = (long)gridDim.x * blockDim.x;
  for (; i < n; i += stride) p[i] = 0.0f;
}

// ---------------------------------------------------------------------------
// Weighted edge scatter: out[dst] += w * x[src].  One wave32 per edge; each
// lane owns 2 consecutive features (64 feats / 32 lanes).  Gathers are
// float2-vectorized; accumulation via fp32 global atomics (L2-resident).
// ---------------------------------------------------------------------------
__global__ void gcn_scatter_kernel(const float* __restrict__ x,
                                   const int* __restrict__ src,
                                   const int* __restrict__ dst,
                                   const float* __restrict__ w,
                                   float* __restrict__ out, int E) {
  int tid  = blockIdx.x * blockDim.x + threadIdx.x;
  int e    = tid >> 5;
  int lane = tid & 31;
  if (e >= E) return;
  int s  = src[e];
  int d  = dst[e];
  float wt = w[e];
  int f = lane * 2;
  const float2 xv = *(const float2*)(x + (long)s * GCN_F + f);
  atomicAdd(out + (long)d * GCN_F + f + 0, wt * xv.x);
  atomicAdd(out + (long)d * GCN_F + f + 1, wt * xv.y);
}

// ---------------------------------------------------------------------------
// WMMA fp32 GEMM: C[n_rows x NC] = A[n_rows x 64] @ W[64 x NC] + bias,
// optional fused leaky-ReLU.  One wave computes a 16-row tile (all NC cols)
// via V_WMMA_F32_16X16X4_F32, K=64 -> 16 steps.
//
// W is staged into LDS pre-packed by K-pairs:
//   Wp[p][col] = { W[2p][col], W[2p+1][col] },  p = 0..31
// so each lane's B fragment (K = {kb,kb+1} at column n) is ONE contiguous
// ds_load_b64 into an even-aligned VGPR pair — no repacking movs.
//
// ISA 7.12.2 32-bit layouts (wave32):
//   A 16x4 : lane&15 = M, VGPR0/1 = K = kb,kb+1 with kb = (lane>>4)*2
//   B 4x16 : lane&15 = N, VGPR0/1 = K = kb,kb+1
//   C 16x16: VGPR r -> M = r + 8*(lane>>4), N = lane&15
// ---------------------------------------------------------------------------
template <int NC, bool RELU>
__global__ void gcn_wmma_gemm_kernel(const float* __restrict__ A,
                                     const float* __restrict__ W,
                                     const float* __restrict__ bias,
                                     float* __restrict__ C, int n_rows) {
  __shared__ float2 Wp[32 * NC];  // K-pair-packed weights
  for (int i = threadIdx.x; i < 32 * NC; i += blockDim.x) {
    const int p = i / NC;
    const int col = i % NC;
    Wp[i] = make_float2(W[(2 * p + 0) * NC + col], W[(2 * p + 1) * NC + col]);
  }
  __syncthreads();

  const int wave = threadIdx.x >> 5;
  const int lane = threadIdx.x & 31;
  const int tile = blockIdx.x * (blockDim.x >> 5) + wave;
  if (tile * 16 >= n_rows) return;   // wave-uniform: EXEC stays all-ones

  const int m    = lane & 15;        // row within tile (A), col within tile (B/C)
  const int half = lane >> 4;        // which half-wave (K sub-offset)
  const int kb   = half * 2;
  const float* __restrict__ Arow = A + (long)(tile * 16 + m) * 64;

  v8f acc[NC / 16];
#pragma unroll
  for (int ct = 0; ct < NC / 16; ++ct) acc[ct] = (v8f){0, 0, 0, 0, 0, 0, 0, 0};

#pragma unroll
  for (int kk = 0; kk < 16; ++kk) {
    const int k0 = kk * 4 + kb;          // this lane's K pair in A
    const int p  = kk * 2 + half;        // this lane's K-pair index in Wp
    v2f a;
    a.x = Arow[k0 + 0];
    a.y = Arow[k0 + 1];
    const float2* __restrict__ Wrow = Wp + p * NC + m;
#pragma unroll
    for (int ct = 0; ct < NC / 16; ++ct) {
      const float2 bv = Wrow[ct * 16];   // single ds_load_b64, imm offset
      v2f b;
      b.x = bv.x;
      b.y = bv.y;
      acc[ct] = __builtin_amdgcn_wmma_f32_16x16x4_f32(
          /*neg_a=*/false, a, /*neg_b=*/false, b,
          /*c_mod=*/(short)0, acc[ct], /*reuse_a=*/false, /*reuse_b=*/false);
    }
  }

  const int mhalf = half * 8;
#pragma unroll
  for (int ct = 0; ct < NC / 16; ++ct) {
    const int col = ct * 16 + m;
    const float bv = bias[col];
#pragma unroll
    for (int r = 0; r < 8; ++r) {
      float v = acc[ct][r] + bv;
      if (RELU) v = (v >= 0.0f) ? v : 0.01f * v;
      C[(long)(tile * 16 + mhalf + r) * NC + col] = v;
    }
  }
}

// ---------------------------------------------------------------------------
// Launch: zero -> scatter(feats)->agg1 -> GEMM1(+bias,+lrelu)->h ->
//         scatter(h)->agg2 -> GEMM2(+bias)->d_out
// ---------------------------------------------------------------------------
extern "C" void kernel_launch(void* const* d_in, const int* in_sizes, int n_in,
                              void* d_out, int out_size, void* d_ws, size_t ws_size,
                              hipStream_t stream) {
  (void)in_sizes; (void)n_in; (void)out_size; (void)ws_size;

  const float* feats = (const float*)d_in[0];
  const int*   esrc  = (const int*)d_in[1];
  const int*   edst  = (const int*)d_in[2];
  const float* ew    = (const float*)d_in[3];
  const float* W1    = (const float*)d_in[4];
  const float* b1    = (const float*)d_in[5];
  const float* W2    = (const float*)d_in[6];
  const float* b2    = (const float*)d_in[7];
  float*       out   = (float*)d_out;

  const long nf = (long)GCN_N * GCN_F;           // 6.4M floats per buffer
  float* agg1 = (float*)d_ws;                    // [N,64]
  float* h    = agg1 + nf;                       // [N,64]
  float* agg2 = h + nf;                          // [N,64]

  // 1) zero agg1, h, agg2 (contiguous 3*nf region; h zeroing is harmless)
  gcn_zero_kernel<<<2048, 256, 0, stream>>>(agg1, 3 * nf);

  // 2) agg1 = A @ feats
  {
    long threads = (long)GCN_E * 32;
    int blocks = (int)((threads + 255) / 256);
    gcn_scatter_kernel<<<blocks, 256, 0, stream>>>(feats, esrc, edst, ew, agg1, GCN_E);
  }

  // 3) h = leaky_relu(agg1 @ W1 + b1)
  {
    int tiles = (GCN_N + 15) / 16;               // 6250
    int blocks = (tiles + 7) / 8;                // 8 waves/block
    gcn_wmma_gemm_kernel<64, true><<<blocks, 256, 0, stream>>>(agg1, W1, b1, h, GCN_N);
  }

  // 4) agg2 = A @ h
  {
    long threads = (long)GCN_E * 32;
    int blocks = (int)((threads + 255) / 256);
    gcn_scatter_kernel<<<blocks, 256, 0, stream>>>(h, esrc, edst, ew, agg2, GCN_E);
  }

  // 5) out = agg2 @ W2 + b2
  {
    int tiles = (GCN_N + 15) / 16;
    int blocks = (tiles + 7) / 8;
    gcn_wmma_gemm_kernel<32, false><<<blocks, 256, 0, stream>>>(agg2, W2, b2, out, GCN_N);
  }
}